// MAM_70849780515288
// MI455X (gfx1250) — compile-verified
//
#include <hip/hip_runtime.h>
#include <hip/hip_bf16.h>

// Reference collapses: the loop's attention math is dead (cur is overwritten by
// a loop-invariant expression at the end of every iteration). Output is exactly
//   out[n,d] = sum_v b_score[n,v] * W2[d,v] + q[n,d]
// i.e. C[4096,512] = b_score[4096,512] @ W2^T[512,512] + q, all fp32.
// Implemented with V_WMMA_F32_16X16X4_F32 (fp32 WMMA, 16x16 tiles, K=4/step).

typedef __attribute__((ext_vector_type(2))) float v2f;
typedef __attribute__((ext_vector_type(8))) float v8f;

#define NROWS 4096
#define DIM   512   // D == V == 512

__global__ __launch_bounds__(256) void MAM_gemm_bias_wmma(
    const float* __restrict__ b_score,  // [4096,512]  (A, row-major)
    const float* __restrict__ W2,       // [512,512]   row-major [d, v]; used as B^T
    const float* __restrict__ q,        // [4096,512]  residual
    float* __restrict__ out)            // [4096,512]
{
    const int lane   = threadIdx.x & 31;
    const int wave   = threadIdx.x >> 5;
    const int mBase  = blockIdx.x * 32;   // each block: 32 rows of C
    const int nBase  = wave * 64;         // each wave: 64 cols (block covers all 512)

    const int laneLo = lane & 15;         // M (A) / N (B) index within tile
    const int laneHi = lane >> 4;         // selects K pair {0,1} vs {2,3}

    // A fragment (16x4 fp32): lane holds A[M = laneLo][K = k + 2*laneHi + {0,1}]
    const float* aPtr0 = b_score + (size_t)(mBase + laneLo) * DIM + 2 * laneHi;
    const float* aPtr1 = aPtr0 + (size_t)16 * DIM;  // second M-tile (M+16)

    // B fragment (4x16 fp32): lane holds B[K = k + 2*laneHi + {0,1}][N = laneLo]
    // B[k][n] = W2[n_global][k_global]  -> row-major read of W2.
    const float* bPtr0 = W2 + (size_t)(nBase +  0 + laneLo) * DIM + 2 * laneHi;
    const float* bPtr1 = W2 + (size_t)(nBase + 16 + laneLo) * DIM + 2 * laneHi;
    const float* bPtr2 = W2 + (size_t)(nBase + 32 + laneLo) * DIM + 2 * laneHi;
    const float* bPtr3 = W2 + (size_t)(nBase + 48 + laneLo) * DIM + 2 * laneHi;

    v8f acc[2][4];
#pragma unroll
    for (int i = 0; i < 2; ++i)
#pragma unroll
        for (int j = 0; j < 4; ++j)
            acc[i][j] = (v8f)(0.0f);

#pragma unroll 2
    for (int k = 0; k < DIM; k += 4) {
        v2f a0 = *(const v2f*)(aPtr0 + k);
        v2f a1 = *(const v2f*)(aPtr1 + k);
        v2f b0 = *(const v2f*)(bPtr0 + k);
        v2f b1 = *(const v2f*)(bPtr1 + k);
        v2f b2 = *(const v2f*)(bPtr2 + k);
        v2f b3 = *(const v2f*)(bPtr3 + k);

        // 8 args: (neg_a, A, neg_b, B, c_mod, C, reuse_a, reuse_b)
        acc[0][0] = __builtin_amdgcn_wmma_f32_16x16x4_f32(false, a0, false, b0, (short)0, acc[0][0], false, false);
        acc[0][1] = __builtin_amdgcn_wmma_f32_16x16x4_f32(false, a0, false, b1, (short)0, acc[0][1], false, false);
        acc[0][2] = __builtin_amdgcn_wmma_f32_16x16x4_f32(false, a0, false, b2, (short)0, acc[0][2], false, false);
        acc[0][3] = __builtin_amdgcn_wmma_f32_16x16x4_f32(false, a0, false, b3, (short)0, acc[0][3], false, false);
        acc[1][0] = __builtin_amdgcn_wmma_f32_16x16x4_f32(false, a1, false, b0, (short)0, acc[1][0], false, false);
        acc[1][1] = __builtin_amdgcn_wmma_f32_16x16x4_f32(false, a1, false, b1, (short)0, acc[1][1], false, false);
        acc[1][2] = __builtin_amdgcn_wmma_f32_16x16x4_f32(false, a1, false, b2, (short)0, acc[1][2], false, false);
        acc[1][3] = __builtin_amdgcn_wmma_f32_16x16x4_f32(false, a1, false, b3, (short)0, acc[1][3], false, false);
    }

    // C/D layout: VGPR r -> M = 8*laneHi + r, N = laneLo. Fuse residual add.
#pragma unroll
    for (int i = 0; i < 2; ++i) {
        const int mOff = mBase + i * 16 + laneHi * 8;
#pragma unroll
        for (int j = 0; j < 4; ++j) {
            const int nOff = nBase + j * 16 + laneLo;
#pragma unroll
            for (int r = 0; r < 8; ++r) {
                const size_t idx = (size_t)(mOff + r) * DIM + nOff;
                out[idx] = acc[i][j][r] + q[idx];
            }
        }
    }
}

extern "C" void kernel_launch(void* const* d_in, const int* in_sizes, int n_in,
                              void* d_out, int out_size, void* d_ws, size_t ws_size,
                              hipStream_t stream) {
    // setup_inputs order: q, k, v, b_score, W1, W2  (k, v, W1 are dead)
    const float* q       = (const float*)d_in[0];
    const float* b_score = (const float*)d_in[3];
    const float* W2      = (const float*)d_in[5];
    float* out = (float*)d_out;

    dim3 grid(NROWS / 32);   // 128 blocks x 256 threads = 1024 waves
    dim3 block(256);
    MAM_gemm_bias_wmma<<<grid, block, 0, stream>>>(b_score, W2, q, out);
}